// DetectorCriterion_90271622627842
// MI455X (gfx1250) — compile-verified
//
#include <hip/hip_runtime.h>
#include <cstdint>

// Problem constants (from reference setup_inputs)
constexpr int B = 32;
constexpr int P = 32768;
constexpr int N = 32;
constexpr int C = 21;
constexpr int BLK = 256;
constexpr float IOU_THRESH = 0.5f;

// ---------------------------------------------------------------------------
// CDNA5 async global->LDS path (gfx1250). Guarded so the file compiles on both
// toolchains and in the host pass; fallback is plain LDS staging.
// Probe result (round 1 diagnostics): the builtins take typed GENERIC pointers:
//   __builtin_amdgcn_global_load_async_to_lds_b128(int4* gsrc, int4* ldst, Ii, Ii)
//   __builtin_amdgcn_global_load_async_to_lds_b32 (int*  gsrc, int*  ldst, Ii, Ii)
// ---------------------------------------------------------------------------
#if defined(__gfx1250__) && \
    __has_builtin(__builtin_amdgcn_global_load_async_to_lds_b128) && \
    __has_builtin(__builtin_amdgcn_global_load_async_to_lds_b32)
#define USE_ASYNC_LDS 1
#else
#define USE_ASYNC_LDS 0
#endif

typedef int v4i __attribute__((vector_size(4 * sizeof(int))));

__device__ __forceinline__ void wait_async_and_barrier() {
#if USE_ASYNC_LDS
#if __has_builtin(__builtin_amdgcn_s_wait_asynccnt)
  __builtin_amdgcn_s_wait_asynccnt(0);
#else
  asm volatile("s_wait_asynccnt 0" ::: "memory");
#endif
#endif
  __syncthreads();
}

// ---------------------------------------------------------------------------
// Kernel 1: best prior per (image, object): first-occurrence argmax over P
// of IoU(prior p, box n). Invalid objects (label<=0) -> 0 (unused later).
// ---------------------------------------------------------------------------
__global__ void __launch_bounds__(BLK) bestprior_kernel(
    const float4* __restrict__ priors,     // [P]
    const float4* __restrict__ gt_boxes,   // [B*N]
    const int* __restrict__ gt_labels,     // [B*N]
    int* __restrict__ best_prior)          // [B*N]
{
  const int idx = blockIdx.x;              // b*N + n
  const int tid = threadIdx.x;
  const int lab = gt_labels[idx];
  if (lab <= 0) { if (tid == 0) best_prior[idx] = 0; return; }

  const float4 g = gt_boxes[idx];
  const float area_b = fmaxf(g.z - g.x, 0.f) * fmaxf(g.w - g.y, 0.f);

  float best = -1.0f;
  int bestp = P;
  for (int p = tid; p < P; p += BLK) {
    const float4 a = priors[p];
    const float ltx = fmaxf(a.x, g.x), lty = fmaxf(a.y, g.y);
    const float rbx = fminf(a.z, g.z), rby = fminf(a.w, g.w);
    const float w = fmaxf(rbx - ltx, 0.f), h = fmaxf(rby - lty, 0.f);
    const float inter = w * h;
    const float area_a = fmaxf(a.z - a.x, 0.f) * fmaxf(a.w - a.y, 0.f);
    const float uni = area_a + area_b - inter;
    const float iou = inter / fmaxf(uni, 1e-10f);
    if (iou > best) { best = iou; bestp = p; }  // strict > keeps first max
  }

  __shared__ float sv[BLK];
  __shared__ int   si[BLK];
  sv[tid] = best; si[tid] = bestp;
  __syncthreads();
  for (int s = BLK / 2; s > 0; s >>= 1) {
    if (tid < s) {
      const float ov = sv[tid + s]; const int oi = si[tid + s];
      if (ov > sv[tid] || (ov == sv[tid] && oi < si[tid])) { sv[tid] = ov; si[tid] = oi; }
    }
    __syncthreads();
  }
  if (tid == 0) best_prior[idx] = si[0];
}

// ---------------------------------------------------------------------------
// Kernel 2: per-(image, prior) matching + NLL + positive losses.
// Stages the 256x21 clf tile and per-image metadata into LDS via CDNA5
// async global->LDS b128/b32 ops (coalesced; LDS reads are conflict-free:
// row stride 21 dwords, gcd(21,64)=1).
// ---------------------------------------------------------------------------
__global__ void __launch_bounds__(BLK) match_kernel(
    const float4* __restrict__ priors,     // [P]
    const float4* __restrict__ gt_boxes,   // [B*N]
    const int* __restrict__ gt_labels,     // [B*N]
    const float* __restrict__ clf,         // [B*P*C]
    const float4* __restrict__ reg,        // [B*P]
    const int* __restrict__ best_prior,    // [B*N]
    float* __restrict__ negnll,            // [B*P]
    int* __restrict__ npos,                // [B]
    double* __restrict__ pos_sum,          // [B]
    double* __restrict__ reg_sum)          // [B]
{
  const int b   = blockIdx.y;
  const int tid = threadIdx.x;
  const int p   = blockIdx.x * BLK + tid;

  __shared__ float  s_clf[BLK * C];        // 21504 B
  __shared__ float4 s_box[N];
  __shared__ int    s_lab[N];
  __shared__ int    s_bp[N];

  const size_t tile_row0 = (size_t)b * P + (size_t)blockIdx.x * BLK;

#if USE_ASYNC_LDS
  {
    const float4* csrc = (const float4*)(clf + tile_row0 * C);
    float4* cdst = (float4*)s_clf;
    for (int i = tid; i < BLK * C / 4; i += BLK)
      __builtin_amdgcn_global_load_async_to_lds_b128((v4i*)(csrc + i), (v4i*)(cdst + i), 0, 0);
    if (tid < N) {
      __builtin_amdgcn_global_load_async_to_lds_b128((v4i*)(gt_boxes + b * N + tid),
                                                     (v4i*)(&s_box[tid]), 0, 0);
      __builtin_amdgcn_global_load_async_to_lds_b32((int*)(gt_labels + b * N + tid),
                                                    (int*)(&s_lab[tid]), 0, 0);
      __builtin_amdgcn_global_load_async_to_lds_b32((int*)(best_prior + b * N + tid),
                                                    (int*)(&s_bp[tid]), 0, 0);
    }
  }
#else
  {
    const float4* csrc = (const float4*)(clf + tile_row0 * C);
    float4* cdst = (float4*)s_clf;
    for (int i = tid; i < BLK * C / 4; i += BLK) cdst[i] = csrc[i];
    if (tid < N) {
      s_box[tid] = gt_boxes[b * N + tid];
      s_lab[tid] = gt_labels[b * N + tid];
      s_bp[tid]  = best_prior[b * N + tid];
    }
  }
#endif
  wait_async_and_barrier();

  // ---- matching: corrected IoU argmax over N (first-occurrence) ----
  const float4 a = priors[p];
  const float area_a = fmaxf(a.z - a.x, 0.f) * fmaxf(a.w - a.y, 0.f);
  float best = -2.0f;
  int bestn = 0;
  for (int n = 0; n < N; ++n) {
    float val;
    if (s_lab[n] > 0) {
      if (p == s_bp[n]) {
        val = 1.0f;                        // forced best-prior match
      } else {
        const float4 g = s_box[n];
        const float ltx = fmaxf(a.x, g.x), lty = fmaxf(a.y, g.y);
        const float rbx = fminf(a.z, g.z), rby = fminf(a.w, g.w);
        const float w = fmaxf(rbx - ltx, 0.f), h = fmaxf(rby - lty, 0.f);
        const float inter = w * h;
        const float area_b = fmaxf(g.z - g.x, 0.f) * fmaxf(g.w - g.y, 0.f);
        const float uni = area_a + area_b - inter;
        val = inter / fmaxf(uni, 1e-10f);
      }
    } else {
      val = -1.0f;                         // invalid (padded) object
    }
    if (val > best) { best = val; bestn = n; }
  }
  const int  tgt = (best >= IOU_THRESH) ? s_lab[bestn] : 0;
  const bool pos = tgt > 0;

  // ---- NLL = logsumexp(row) - row[tgt], from LDS ----
  const float* row = s_clf + tid * C;
  float m = row[0];
#pragma unroll
  for (int c = 1; c < C; ++c) m = fmaxf(m, row[c]);
  float s = 0.f;
#pragma unroll
  for (int c = 0; c < C; ++c) s += expf(row[c] - m);
  const float nll = m + logf(s) - row[tgt];

  negnll[(size_t)b * P + p] = pos ? -1.0f : nll;   // sentinel for positives

  float rs = 0.f;
  if (pos) {
    const float4 r = reg[(size_t)b * P + p];
    const float4 g = s_box[bestn];
    const float dx = r.x - g.x, dy = r.y - g.y, dz = r.z - g.z, dw = r.w - g.w;
    rs = dx * dx + dy * dy + dz * dz + dw * dw;
  }

  // ---- block reduce (count, pos-NLL sum, reg-MSE sum) ----
  __shared__ float rf1[BLK];
  __shared__ float rf2[BLK];
  __shared__ int   ri[BLK];
  rf1[tid] = pos ? nll : 0.f;
  rf2[tid] = rs;
  ri[tid]  = pos ? 1 : 0;
  __syncthreads();
  for (int s2 = BLK / 2; s2 > 0; s2 >>= 1) {
    if (tid < s2) {
      rf1[tid] += rf1[tid + s2];
      rf2[tid] += rf2[tid + s2];
      ri[tid]  += ri[tid + s2];
    }
    __syncthreads();
  }
  if (tid == 0) {
    if (ri[0]) atomicAdd(&npos[b], ri[0]);
    atomicAdd(&pos_sum[b], (double)rf1[0]);
    atomicAdd(&reg_sum[b], (double)rf2[0]);
  }
}

// ---------------------------------------------------------------------------
// Kernel 3: OHEM top-k sum per image via 4x8-bit radix select on float bits
// (all eligible values >= 0, so uint bit order == float order). Exact under
// ties: sum(v > T) + kk * T.
// ---------------------------------------------------------------------------
__global__ void __launch_bounds__(BLK) ohem_kernel(
    const float* __restrict__ negnll,      // [B*P]
    const int* __restrict__ npos,          // [B]
    double* __restrict__ neg_sum)          // [B]
{
  const int b = blockIdx.x;
  const int tid = threadIdx.x;
  const float* v = negnll + (size_t)b * P;

  const int np = npos[b];
  const int k = 3 * np;
  if (k <= 0) { if (tid == 0) neg_sum[b] = 0.0; return; }
  const int elig = P - np;
  const bool take_all = (k >= elig);

  __shared__ int      hist[256];
  __shared__ unsigned s_prefix;
  __shared__ int      s_kk;
  __shared__ double   sred[BLK];

  unsigned T = 0;
  int kk = 0;
  if (!take_all) {
    if (tid == 0) { s_prefix = 0u; s_kk = k; }
    for (int pass = 0; pass < 4; ++pass) {
      const int shift = 24 - 8 * pass;
      const unsigned himask = (pass == 0) ? 0u : (0xFFFFFFFFu << (shift + 8));
      hist[tid] = 0;
      __syncthreads();
      const unsigned pref = s_prefix;
      for (int p = tid; p < P; p += BLK) {
        const float x = v[p];
        if (x >= 0.0f) {
          const unsigned u = __float_as_uint(x);
          if ((u & himask) == pref) atomicAdd(&hist[(u >> shift) & 0xFF], 1);
        }
      }
      __syncthreads();
      if (tid == 0) {
        int kk0 = s_kk;
        int d = 255;
        for (; d > 0; --d) {
          const int c = hist[d];
          if (kk0 - c <= 0) break;
          kk0 -= c;
        }
        s_prefix = pref | ((unsigned)d << shift);
        s_kk = kk0;
      }
      __syncthreads();
    }
    T = s_prefix;
    kk = s_kk;
  }

  double local = 0.0;
  for (int p = tid; p < P; p += BLK) {
    const float x = v[p];
    if (x >= 0.0f) {
      if (take_all) local += (double)x;
      else if (__float_as_uint(x) > T) local += (double)x;
    }
  }
  sred[tid] = local;
  __syncthreads();
  for (int s = BLK / 2; s > 0; s >>= 1) {
    if (tid < s) sred[tid] += sred[tid + s];
    __syncthreads();
  }
  if (tid == 0) {
    double tot = sred[0];
    if (!take_all) tot += (double)kk * (double)__uint_as_float(T);
    neg_sum[b] = tot;
  }
}

// ---------------------------------------------------------------------------
// Kernel 4: final loss = mean_b(pos/npos + neg/max(3*npos,1) + reg/(npos*4))
// ---------------------------------------------------------------------------
__global__ void finalize_kernel(
    const int* __restrict__ npos,
    const double* __restrict__ pos_sum,
    const double* __restrict__ reg_sum,
    const double* __restrict__ neg_sum,
    float* __restrict__ out)
{
  const int b = threadIdx.x;   // 32 threads
  __shared__ double s[B];
  const int np = npos[b];
  const double npf = (double)(np > 0 ? np : 1);
  const double nnf = (double)(np > 0 ? 3 * np : 1);
  const double clf_l = pos_sum[b] / npf + neg_sum[b] / nnf;
  const double reg_l = reg_sum[b] / (npf * 4.0);
  s[b] = clf_l + reg_l;
  __syncthreads();
  if (b == 0) {
    double t = 0.0;
    for (int i = 0; i < B; ++i) t += s[i];
    out[0] = (float)(t / (double)B);
  }
}

// Zero the per-image accumulators each call (deterministic; graph-safe).
__global__ void zero_acc_kernel(double* pos_sum, double* reg_sum,
                                double* neg_sum, int* npos)
{
  const int t = threadIdx.x;
  if (t < B) { pos_sum[t] = 0.0; reg_sum[t] = 0.0; neg_sum[t] = 0.0; npos[t] = 0; }
}

// ---------------------------------------------------------------------------
extern "C" void kernel_launch(void* const* d_in, const int* in_sizes, int n_in,
                              void* d_out, int out_size, void* d_ws, size_t ws_size,
                              hipStream_t stream) {
  (void)in_sizes; (void)n_in; (void)out_size; (void)ws_size;

  const float4* priors   = (const float4*)d_in[0];   // [P,4]
  const float4* gt_boxes = (const float4*)d_in[1];   // [B,N,4]
  const int*    gt_label = (const int*)d_in[2];      // [B,N]
  const float*  clf      = (const float*)d_in[3];    // [B,P,C]
  const float4* reg      = (const float4*)d_in[4];   // [B,P,4]
  float*        out      = (float*)d_out;

  // Workspace layout: 4 MiB negnll + small per-image accumulators.
  char* ws = (char*)d_ws;
  float*  negnll  = (float*)ws;
  double* pos_sum = (double*)(ws + (size_t)B * P * sizeof(float));
  double* reg_sum = pos_sum + B;
  double* neg_sum = reg_sum + B;
  int*    npos    = (int*)(neg_sum + B);
  int*    bp      = npos + B;

  zero_acc_kernel<<<1, B, 0, stream>>>(pos_sum, reg_sum, neg_sum, npos);

  bestprior_kernel<<<B * N, BLK, 0, stream>>>(priors, gt_boxes, gt_label, bp);

  dim3 grid2(P / BLK, B);
  match_kernel<<<grid2, BLK, 0, stream>>>(priors, gt_boxes, gt_label, clf, reg,
                                          bp, negnll, npos, pos_sum, reg_sum);

  ohem_kernel<<<B, BLK, 0, stream>>>(negnll, npos, neg_sum);

  finalize_kernel<<<1, B, 0, stream>>>(npos, pos_sum, reg_sum, neg_sum, out);
}